// SGP_LAYER_78219944395265
// MI455X (gfx1250) — compile-verified
//
#include <hip/hip_runtime.h>
#include <math.h>

constexpr int B = 8, L = 512, H = 8, D = 64, M = 64, S = 2;
constexpr int HD = H * D;     // 512
constexpr int QKVD = 2 * HD;  // 1024
constexpr float JITTER = 1e-4f;

typedef __attribute__((ext_vector_type(2))) float v2f;
typedef __attribute__((ext_vector_type(8))) float v8f;

__device__ __forceinline__ v8f wmma4(v2f a, v2f b, v8f c) {
  // D = A(16x4,f32) * B(4x16,f32) + C(16x16,f32)
  return __builtin_amdgcn_wmma_f32_16x16x4_f32(false, a, false, b, (short)0, c, false, false);
}
__device__ __forceinline__ v8f zero8() {
  v8f z = {0.f, 0.f, 0.f, 0.f, 0.f, 0.f, 0.f, 0.f};
  return z;
}

// ---------------- 1. qkv = x @ W_qkv^T, split into q[(b,h,l,d)] and vg[(b,h,l,d)]
__global__ __launch_bounds__(128) void k_qkv(const float* __restrict__ x,
                                             const float* __restrict__ Wqkv,
                                             float* __restrict__ q, float* __restrict__ vg) {
  int wave = blockIdx.x * 4 + (threadIdx.x >> 5);
  int lane = threadIdx.x & 31, half = lane >> 4, lan = lane & 15;
  int rt = wave >> 6, ct = wave & 63;  // 256 row tiles (B*L/16), 64 col tiles (1024/16)
  int row0 = rt * 16, col0 = ct * 16;
  v8f acc = zero8();
  const float* Ar = x + (size_t)(row0 + lan) * HD;
  const float* Bc = Wqkv + (size_t)(col0 + lan) * HD;  // B[k][n] = Wqkv[n][k]
  for (int k0 = 0; k0 < HD; k0 += 4) {
    int kb = k0 + 2 * half;
    v2f a; a.x = Ar[kb]; a.y = Ar[kb + 1];
    v2f b; b.x = Bc[kb]; b.y = Bc[kb + 1];
    acc = wmma4(a, b, acc);
  }
#pragma unroll
  for (int r = 0; r < 8; ++r) {
    int row = row0 + r + 8 * half;
    int c = col0 + lan;
    int bb = row / L, l = row % L;
    int h = c >> 7, rem = c & 127;
    float v = acc[r];
    size_t o = ((size_t)(bb * H + h) * L + l) * D;
    if (rem < D) q[o + rem] = v; else vg[o + rem - D] = v;
  }
}

// ---------------- 2. k_beta[h,m,d] = sum_k Wqkv[h*64+d][k]*cur_k[m][k]
__global__ __launch_bounds__(128) void k_kbeta(const float* __restrict__ curk,
                                               const float* __restrict__ Wqkv,
                                               float* __restrict__ kb) {
  int wave = blockIdx.x * 4 + (threadIdx.x >> 5);
  int lane = threadIdx.x & 31, half = lane >> 4, lan = lane & 15;
  int rt = wave >> 5, ct = wave & 31;  // rows M=64 -> 4 tiles, cols 512 -> 32 tiles
  int row0 = rt * 16, col0 = ct * 16;
  v8f acc = zero8();
  const float* Ar = curk + (size_t)(row0 + lan) * HD;
  const float* Bc = Wqkv + (size_t)(col0 + lan) * HD;
  for (int k0 = 0; k0 < HD; k0 += 4) {
    int kbx = k0 + 2 * half;
    v2f a; a.x = Ar[kbx]; a.y = Ar[kbx + 1];
    v2f b; b.x = Bc[kbx]; b.y = Bc[kbx + 1];
    acc = wmma4(a, b, acc);
  }
#pragma unroll
  for (int r = 0; r < 8; ++r) {
    int m = row0 + r + 8 * half;
    int c = col0 + lan;
    int h = c >> 6, d = c & 63;
    kb[((size_t)h * M + m) * D + d] = acc[r];
  }
}

// ---------------- 3a. inv_ls = exp(-log_ls); sf = exp(log_sf)
__global__ void k_expp(const float* __restrict__ log_ls, const float* __restrict__ log_sf,
                       float* __restrict__ invls, float* __restrict__ sfv) {
  int i = threadIdx.x;
  if (i < H * D) invls[i] = expf(-log_ls[i]);
  if (i < H) sfv[i] = expf(log_sf[i]);
}

// ---------------- 3b. in-place scale q *= inv_ls; a2[b,h,l] = |q_scaled|^2
__global__ __launch_bounds__(64) void k_scaleq(float* __restrict__ q,
                                               const float* __restrict__ invls,
                                               float* __restrict__ a2) {
  int idx = blockIdx.x;              // bh*L + l
  int h = (idx / L) % H;
  int d = threadIdx.x;
  float v = q[(size_t)idx * D + d] * invls[h * D + d];
  q[(size_t)idx * D + d] = v;
  __shared__ float red[64];
  red[d] = v * v;
  __syncthreads();
  if (d == 0) {
    float s = 0.f;
    for (int i = 0; i < 64; ++i) s += red[i];
    a2[idx] = s;
  }
}

// ---------------- 3c. in-place scale k_beta; bk2[h,m]
__global__ __launch_bounds__(64) void k_scalek(float* __restrict__ kb,
                                               const float* __restrict__ invls,
                                               float* __restrict__ bk2) {
  int idx = blockIdx.x;  // h*M+m
  int h = idx / M;
  int d = threadIdx.x;
  float v = kb[(size_t)idx * D + d] * invls[h * D + d];
  kb[(size_t)idx * D + d] = v;
  __shared__ float red[64];
  red[d] = v * v;
  __syncthreads();
  if (d == 0) {
    float s = 0.f;
    for (int i = 0; i < 64; ++i) s += red[i];
    bk2[idx] = s;
  }
}

// ---------------- 4. K_kk[h] = sf*exp(-0.5*(bk2[m]+bk2[n]-2 bk[m].bk[n]))
__global__ __launch_bounds__(128) void k_kkk(const float* __restrict__ bk,
                                             const float* __restrict__ bk2,
                                             const float* __restrict__ sfv,
                                             float* __restrict__ kkk) {
  int wave = blockIdx.x * 4 + (threadIdx.x >> 5);
  int lane = threadIdx.x & 31, half = lane >> 4, lan = lane & 15;
  int h = wave >> 4, t = wave & 15, rt = t >> 2, ct = t & 3;
  int row0 = rt * 16, col0 = ct * 16;
  v8f acc = zero8();
  const float* Ar = bk + ((size_t)h * M + row0 + lan) * D;
  const float* Bc = bk + ((size_t)h * M + col0 + lan) * D;
  for (int k0 = 0; k0 < D; k0 += 4) {
    int kbx = k0 + 2 * half;
    v2f a; a.x = Ar[kbx]; a.y = Ar[kbx + 1];
    v2f b; b.x = Bc[kbx]; b.y = Bc[kbx + 1];
    acc = wmma4(a, b, acc);
  }
  float sf = sfv[h];
#pragma unroll
  for (int r = 0; r < 8; ++r) {
    int row = row0 + r + 8 * half, col = col0 + lan;
    float d2 = bk2[h * M + row] + bk2[h * M + col] - 2.f * acc[r];
    kkk[(size_t)h * M * M + row * M + col] = sf * expf(-0.5f * d2);
  }
}

// ---------------- 5. Cholesky of K_kk + JITTER*I (per head, left-looking in LDS)
__global__ __launch_bounds__(64) void k_chol(const float* __restrict__ kkk,
                                             float* __restrict__ chol) {
  __shared__ float A[64][64];
  int h = blockIdx.x, t = threadIdx.x;
  for (int j = 0; j < 64; ++j)
    A[t][j] = kkk[(size_t)h * M * M + t * M + j] + (t == j ? JITTER : 0.f);
  __syncthreads();
  for (int j = 0; j < 64; ++j) {
    if (t == j) {
      float s = A[j][j];
      for (int k = 0; k < j; ++k) s -= A[j][k] * A[j][k];
      A[j][j] = sqrtf(s);
    }
    __syncthreads();
    if (t > j) {
      float s = A[t][j];
      for (int k = 0; k < j; ++k) s -= A[t][k] * A[j][k];
      A[t][j] = s / A[j][j];
    }
    __syncthreads();
  }
  for (int j = 0; j < 64; ++j)
    chol[(size_t)h * M * M + t * M + j] = (j <= t) ? A[t][j] : 0.f;
}

// ---------------- 6. K_qk_beta[b,h,l,m]
__global__ __launch_bounds__(128) void k_kqkb(const float* __restrict__ as,
                                              const float* __restrict__ a2,
                                              const float* __restrict__ bk,
                                              const float* __restrict__ bk2,
                                              const float* __restrict__ sfv,
                                              float* __restrict__ kqk) {
  int wave = blockIdx.x * 4 + (threadIdx.x >> 5);
  int lane = threadIdx.x & 31, half = lane >> 4, lan = lane & 15;
  int bh = wave >> 7, t = wave & 127, lt = t >> 2, mt = t & 3;
  int h = bh % H;
  int l0 = lt * 16, m0 = mt * 16;
  const float* Ar = as + ((size_t)bh * L + l0 + lan) * D;
  const float* Bc = bk + ((size_t)h * M + m0 + lan) * D;
  v8f acc = zero8();
  for (int k0 = 0; k0 < D; k0 += 4) {
    int kbx = k0 + 2 * half;
    v2f a; a.x = Ar[kbx]; a.y = Ar[kbx + 1];
    v2f b; b.x = Bc[kbx]; b.y = Bc[kbx + 1];
    acc = wmma4(a, b, acc);
  }
  float sf = sfv[h];
#pragma unroll
  for (int r = 0; r < 8; ++r) {
    int row = l0 + r + 8 * half, col = m0 + lan;
    float d2 = a2[(size_t)bh * L + row] + bk2[h * M + col] - 2.f * acc[r];
    kqk[((size_t)bh * L + row) * M + col] = sf * expf(-0.5f * d2);
  }
}

// ---------------- 7. mean1 = K_qq @ v_gamma, fused (flash-style, never materialize K_qq)
__global__ __launch_bounds__(128) void k_mean1(const float* __restrict__ as,
                                               const float* __restrict__ a2,
                                               const float* __restrict__ sfv,
                                               const float* __restrict__ vg,
                                               float* __restrict__ mean1) {
  __shared__ float Ks[4][16][17];
  int w = threadIdx.x >> 5;
  int wave = blockIdx.x * 4 + w;
  int lane = threadIdx.x & 31, half = lane >> 4, lan = lane & 15;
  int bh = wave >> 5, lt = wave & 31;
  int h = bh % H;
  int l0 = lt * 16;
  const float* A = as + (size_t)bh * L * D;
  const float* a2p = a2 + (size_t)bh * L;
  const float* VG = vg + (size_t)bh * L * D;
  float sf = sfv[h];
  v8f macc[4];
#pragma unroll
  for (int i = 0; i < 4; ++i) macc[i] = zero8();
  for (int jt = 0; jt < 32; ++jt) {
    int j0 = jt * 16;
    // kernel tile K(l0.., j0..) via f32 WMMA, K-dim = D
    v8f kac = zero8();
    for (int k0 = 0; k0 < D; k0 += 4) {
      int kbx = k0 + 2 * half;
      v2f a; a.x = A[(size_t)(l0 + lan) * D + kbx]; a.y = A[(size_t)(l0 + lan) * D + kbx + 1];
      v2f b; b.x = A[(size_t)(j0 + lan) * D + kbx]; b.y = A[(size_t)(j0 + lan) * D + kbx + 1];
      kac = wmma4(a, b, kac);
    }
    float a2c = a2p[j0 + lan];
#pragma unroll
    for (int r = 0; r < 8; ++r) {
      int rl = r + 8 * half;
      float d2 = a2p[l0 + rl] + a2c - 2.f * kac[r];
      Ks[w][rl][lan] = sf * expf(-0.5f * d2);  // C-layout -> LDS (in-wave DS ops are ordered)
    }
    // mean1 tile += Ktile(16x16) @ vg_tile(16x64)
#pragma unroll
    for (int ct = 0; ct < 4; ++ct) {
#pragma unroll
      for (int k0 = 0; k0 < 16; k0 += 4) {
        int kbx = k0 + 2 * half;
        v2f a; a.x = Ks[w][lan][kbx]; a.y = Ks[w][lan][kbx + 1];
        v2f b;
        b.x = VG[(size_t)(j0 + kbx) * D + ct * 16 + lan];
        b.y = VG[(size_t)(j0 + kbx + 1) * D + ct * 16 + lan];
        macc[ct] = wmma4(a, b, macc[ct]);
      }
    }
  }
#pragma unroll
  for (int ct = 0; ct < 4; ++ct)
#pragma unroll
    for (int r = 0; r < 8; ++r)
      mean1[((size_t)bh * L + l0 + r + 8 * half) * D + ct * 16 + lan] = macc[ct][r];
}

// ---------------- 8. v1 = chol^-1 K_kq  (forward substitution, thread per column l)
__global__ __launch_bounds__(256) void k_trsv1(const float* __restrict__ chol,
                                               const float* __restrict__ kqk,
                                               float* __restrict__ v1) {
  __shared__ float Lc[64][64];
  int bh = blockIdx.x >> 1, part = blockIdx.x & 1;
  int h = bh % H;
  int t = threadIdx.x;
  for (int i = t; i < 64 * 64; i += 256) Lc[i >> 6][i & 63] = chol[(size_t)h * M * M + i];
  __syncthreads();
  int l = part * 256 + t;
  const float* KQ = kqk + ((size_t)bh * L + l) * M;
  float y[64];
#pragma unroll
  for (int m = 0; m < 64; ++m) {
    float s = KQ[m];
#pragma unroll
    for (int k = 0; k < m; ++k) s -= Lc[m][k] * y[k];
    y[m] = s / Lc[m][m];
    v1[((size_t)bh * M + m) * L + l] = y[m];
  }
}

// ---------------- 9a. rhs2[b,h,m,d] = sum_l kqk[l][m] * vg[l][d]
__global__ __launch_bounds__(128) void k_rhs2(const float* __restrict__ kqk,
                                              const float* __restrict__ vg,
                                              float* __restrict__ rhs2) {
  int wave = blockIdx.x * 4 + (threadIdx.x >> 5);
  int lane = threadIdx.x & 31, half = lane >> 4, lan = lane & 15;
  int bh = wave >> 4, t = wave & 15, mt = t >> 2, dt = t & 3;
  int m0 = mt * 16, d0 = dt * 16;
  const float* KQ = kqk + (size_t)bh * L * M;
  const float* VG = vg + (size_t)bh * L * D;
  v8f acc = zero8();
  for (int k0 = 0; k0 < L; k0 += 4) {
    int kbx = k0 + 2 * half;
    v2f a; a.x = KQ[(size_t)kbx * M + m0 + lan]; a.y = KQ[(size_t)(kbx + 1) * M + m0 + lan];
    v2f b; b.x = VG[(size_t)kbx * D + d0 + lan]; b.y = VG[(size_t)(kbx + 1) * D + d0 + lan];
    acc = wmma4(a, b, acc);
  }
#pragma unroll
  for (int r = 0; r < 8; ++r)
    rhs2[((size_t)bh * M + m0 + r + 8 * half) * D + d0 + lan] = acc[r];
}

// ---------------- 9b. v2 = chol^-1 rhs2
__global__ __launch_bounds__(64) void k_trsv2(const float* __restrict__ chol,
                                              const float* __restrict__ rhs2,
                                              float* __restrict__ v2) {
  __shared__ float Lc[64][64];
  int bh = blockIdx.x;
  int h = bh % H;
  int t = threadIdx.x;
  for (int i = t; i < 64 * 64; i += 64) Lc[i >> 6][i & 63] = chol[(size_t)h * M * M + i];
  __syncthreads();
  int d = t;
  float y[64];
#pragma unroll
  for (int m = 0; m < 64; ++m) {
    float s = rhs2[((size_t)bh * M + m) * D + d];
#pragma unroll
    for (int k = 0; k < m; ++k) s -= Lc[m][k] * y[k];
    y[m] = s / Lc[m][m];
    v2[((size_t)bh * M + m) * D + d] = y[m];
  }
}

// ---------------- 10a. s_local[h,d,m,n] = diag(exp(lsd)) + strict lower tri
__global__ __launch_bounds__(256) void k_slocal(const float* __restrict__ lsd,
                                                const float* __restrict__ ltri,
                                                float* __restrict__ sl) {
  size_t idx = (size_t)blockIdx.x * 256 + threadIdx.x;
  int n = idx & 63, m = (int)(idx >> 6) & 63;
  int hd = (int)(idx >> 12);
  float v;
  if (m == n) v = expf(lsd[(size_t)hd * M + m]);
  else if (m > n) v = ltri[idx];
  else v = 0.f;
  sl[idx] = v;
}

// ---------------- 10b. G[h,d] = s_local[h,d] @ s_local[h,d]^T
__global__ __launch_bounds__(128) void k_G(const float* __restrict__ sl, float* __restrict__ G) {
  int wave = blockIdx.x * 4 + (threadIdx.x >> 5);
  int lane = threadIdx.x & 31, half = lane >> 4, lan = lane & 15;
  int hd = wave >> 4, t = wave & 15, rt = t >> 2, ct = t & 3;
  int m0 = rt * 16, n0 = ct * 16;
  const float* Sp = sl + (size_t)hd * M * M;
  v8f acc = zero8();
  for (int k0 = 0; k0 < M; k0 += 4) {
    int kbx = k0 + 2 * half;
    v2f a; a.x = Sp[(size_t)(m0 + lan) * M + kbx]; a.y = Sp[(size_t)(m0 + lan) * M + kbx + 1];
    v2f b; b.x = Sp[(size_t)(n0 + lan) * M + kbx]; b.y = Sp[(size_t)(n0 + lan) * M + kbx + 1];
    acc = wmma4(a, b, acc);
  }
#pragma unroll
  for (int r = 0; r < 8; ++r)
    G[(size_t)hd * M * M + (m0 + r + 8 * half) * M + n0 + lan] = acc[r];
}

// ---------------- 10c. v3sq[b,h,l,d] = v1(:,l)^T G[h,d] v1(:,l), fused t = G@v1 + reduce
__global__ __launch_bounds__(128) void k_v3sq(const float* __restrict__ G,
                                              const float* __restrict__ v1,
                                              float* __restrict__ v3sq) {
  int wave = blockIdx.x * 4 + (threadIdx.x >> 5);
  int lane = threadIdx.x & 31, half = lane >> 4, lan = lane & 15;
  int lt = wave & 31, d = (wave >> 5) & 63, bh = wave >> 11;
  int h = bh % H;
  int l0 = lt * 16;
  const float* Gp = G + ((size_t)h * D + d) * M * M;
  const float* V1 = v1 + (size_t)bh * M * L;
  float part = 0.f;
#pragma unroll
  for (int mt = 0; mt < 4; ++mt) {
    v8f acc = zero8();
    for (int k0 = 0; k0 < M; k0 += 4) {
      int kbx = k0 + 2 * half;
      v2f a; a.x = Gp[(size_t)(mt * 16 + lan) * M + kbx]; a.y = Gp[(size_t)(mt * 16 + lan) * M + kbx + 1];
      v2f b; b.x = V1[(size_t)kbx * L + l0 + lan]; b.y = V1[(size_t)(kbx + 1) * L + l0 + lan];
      acc = wmma4(a, b, acc);
    }
#pragma unroll
    for (int r = 0; r < 8; ++r) {
      int mrow = mt * 16 + r + 8 * half;
      part += acc[r] * V1[(size_t)mrow * L + l0 + lan];
    }
  }
  part += __shfl_xor(part, 16, 32);
  if (half == 0) v3sq[((size_t)bh * L + l0 + lan) * D + d] = part;
}

// ---------------- 11. v1sq[b,h,l] = sum_m v1[m,l]^2
__global__ __launch_bounds__(256) void k_v1sq(const float* __restrict__ v1,
                                              float* __restrict__ v1sq) {
  int bh = blockIdx.x >> 1, part = blockIdx.x & 1;
  int l = part * 256 + threadIdx.x;
  float acc = 0.f;
  for (int m = 0; m < M; ++m) {
    float v = v1[((size_t)bh * M + m) * L + l];
    acc += v * v;
  }
  v1sq[(size_t)bh * L + l] = acc;
}

// ---------------- 12. mean = mean1 - v1^T v2 + kqk @ v_beta
__global__ __launch_bounds__(128) void k_mean(const float* __restrict__ kqk,
                                              const float* __restrict__ vind,
                                              const float* __restrict__ v1,
                                              const float* __restrict__ v2,
                                              const float* __restrict__ mean1,
                                              float* __restrict__ meanv) {
  int wave = blockIdx.x * 4 + (threadIdx.x >> 5);
  int lane = threadIdx.x & 31, half = lane >> 4, lan = lane & 15;
  int bh = wave >> 7, t = wave & 127, lt = t >> 2, dt = t & 3;
  int h = bh % H;
  int l0 = lt * 16, d0 = dt * 16;
  const float* KQ = kqk + (size_t)bh * L * M;
  const float* VB = vind + (size_t)h * M * D;
  const float* V1 = v1 + (size_t)bh * M * L;
  const float* V2 = v2 + (size_t)bh * M * D;
  v8f acc = zero8();
  for (int k0 = 0; k0 < M; k0 += 4) {  // + kqk @ v_beta
    int kbx = k0 + 2 * half;
    v2f a; a.x = KQ[(size_t)(l0 + lan) * M + kbx]; a.y = KQ[(size_t)(l0 + lan) * M + kbx + 1];
    v2f b; b.x = VB[(size_t)kbx * D + d0 + lan]; b.y = VB[(size_t)(kbx + 1) * D + d0 + lan];
    acc = wmma4(a, b, acc);
  }
  for (int k0 = 0; k0 < M; k0 += 4) {  // - v1^T v2 (negate A on load)
    int kbx = k0 + 2 * half;
    v2f a; a.x = -V1[(size_t)kbx * L + l0 + lan]; a.y = -V1[(size_t)(kbx + 1) * L + l0 + lan];
    v2f b; b.x = V2[(size_t)kbx * D + d0 + lan]; b.y = V2[(size_t)(kbx + 1) * D + d0 + lan];
    acc = wmma4(a, b, acc);
  }
#pragma unroll
  for (int r = 0; r < 8; ++r) {
    size_t o = ((size_t)bh * L + l0 + r + 8 * half) * D + d0 + lan;
    meanv[o] = acc[r] + mean1[o];
  }
}

// ---------------- 13. samples -> (B,S,L,H*D) layout
__global__ __launch_bounds__(256) void k_samples(const float* __restrict__ meanv,
                                                 const float* __restrict__ v3sq,
                                                 const float* __restrict__ v1sq,
                                                 const float* __restrict__ sfv,
                                                 const float* __restrict__ eps,
                                                 float* __restrict__ samp) {
  size_t idx = (size_t)blockIdx.x * 256 + threadIdx.x;  // < B*S*L*HD
  int hd = (int)(idx & (HD - 1));
  int h = hd >> 6, d = hd & 63;
  size_t t = idx >> 9;
  int l = (int)(t & (L - 1)); t >>= 9;
  int s = (int)(t & (S - 1)); t >>= 1;
  int b = (int)t;
  size_t bhl = (size_t)(b * H + h) * L + l;
  float cc = sqrtf(sfv[h] + v3sq[bhl * D + d] - v1sq[bhl]);
  float e = eps[((((size_t)b * H + h) * S + s) * L + l) * D + d];
  samp[idx] = meanv[bhl * D + d] + cc * e;
}

// ---------------- 14. out = samp @ W_O_w^T + W_O_b
__global__ __launch_bounds__(128) void k_out(const float* __restrict__ samp,
                                             const float* __restrict__ WOw,
                                             const float* __restrict__ WOb,
                                             float* __restrict__ out) {
  int wave = blockIdx.x * 4 + (threadIdx.x >> 5);
  int lane = threadIdx.x & 31, half = lane >> 4, lan = lane & 15;
  int rt = wave >> 5, ct = wave & 31;  // rows B*S*L/16=512, cols 512/16=32
  int row0 = rt * 16, col0 = ct * 16;
  const float* Ar = samp + (size_t)(row0 + lan) * HD;
  const float* Bc = WOw + (size_t)(col0 + lan) * HD;
  v8f acc = zero8();
  for (int k0 = 0; k0 < HD; k0 += 4) {
    int kbx = k0 + 2 * half;
    v2f a; a.x = Ar[kbx]; a.y = Ar[kbx + 1];
    v2f b; b.x = Bc[kbx]; b.y = Bc[kbx + 1];
    acc = wmma4(a, b, acc);
  }
#pragma unroll
  for (int r = 0; r < 8; ++r) {
    int row = row0 + r + 8 * half, col = col0 + lan;
    out[(size_t)row * HD + col] = acc[r] + WOb[col];
  }
}

// ---------------- KL reductions (deterministic: single-block, fixed order)
__global__ __launch_bounds__(256) void k_red_sl(const float* __restrict__ sl,
                                                const float* __restrict__ lsd,
                                                float* __restrict__ klp) {
  __shared__ float r0[256], r1[256];
  float s0 = 0.f, s1 = 0.f;
  for (size_t i = threadIdx.x; i < (size_t)H * D * M * M; i += 256) { float v = sl[i]; s0 += v * v; }
  for (int i = threadIdx.x; i < H * D * M; i += 256) s1 += lsd[i];
  r0[threadIdx.x] = s0; r1[threadIdx.x] = s1;
  __syncthreads();
  if (threadIdx.x == 0) {
    float a = 0.f, b = 0.f;
    for (int i = 0; i < 256; ++i) { a += r0[i]; b += r1[i]; }
    klp[0] = a; klp[1] = b;
  }
}

__global__ __launch_bounds__(256) void k_red_vm(const float* __restrict__ vg,
                                                const float* __restrict__ mean1,
                                                const float* __restrict__ v2,
                                                float* __restrict__ klp) {
  __shared__ float r0[256], r1[256];
  float s0 = 0.f, s1 = 0.f;
  for (size_t i = threadIdx.x; i < (size_t)B * H * L * D; i += 256) s0 += vg[i] * mean1[i];
  for (size_t i = threadIdx.x; i < (size_t)B * H * M * D; i += 256) { float v = v2[i]; s1 += v * v; }
  r0[threadIdx.x] = s0; r1[threadIdx.x] = s1;
  __syncthreads();
  if (threadIdx.x == 0) {
    float a = 0.f, b = 0.f;
    for (int i = 0; i < 256; ++i) { a += r0[i]; b += r1[i]; }
    klp[3] = a; klp[4] = b;
  }
}

__global__ __launch_bounds__(64) void k_term3(const float* __restrict__ kkk,
                                              const float* __restrict__ vb,
                                              float* __restrict__ t3p) {
  int h = blockIdx.x, d = threadIdx.x;
  const float* Kh = kkk + (size_t)h * M * M;
  const float* vbh = vb + (size_t)h * M * D;
  float acc = 0.f;
  for (int m = 0; m < M; ++m) {
    float t = 0.f;
    for (int n = 0; n < M; ++n) t += Kh[m * M + n] * vbh[n * D + d];
    acc += vbh[m * D + d] * t;
  }
  __shared__ float r[64];
  r[d] = acc;
  __syncthreads();
  if (d == 0) {
    float s = 0.f;
    for (int i = 0; i < 64; ++i) s += r[i];
    t3p[h] = s;
  }
}

__global__ void k_fin(const float* __restrict__ klp, float* __restrict__ out) {
  float t3 = 0.f;
  for (int h = 0; h < H; ++h) t3 += klp[8 + h];
  float kl = -0.5f * (float)(M * D * H) + 0.5f * klp[0] - klp[1] + 0.5f * t3 +
             (0.5f / (float)B) * (klp[3] - klp[4]);
  out[(size_t)B * S * L * HD] = kl;
}

extern "C" void kernel_launch(void* const* d_in, const int* in_sizes, int n_in,
                              void* d_out, int out_size, void* d_ws, size_t ws_size,
                              hipStream_t stream) {
  (void)in_sizes; (void)n_in; (void)out_size; (void)ws_size;
  const float* x     = (const float*)d_in[0];
  const float* curk  = (const float*)d_in[1];
  const float* eps   = (const float*)d_in[2];
  const float* Wqkv  = (const float*)d_in[3];
  const float* logsf = (const float*)d_in[4];
  const float* logls = (const float*)d_in[5];
  const float* vind  = (const float*)d_in[6];
  const float* ltri  = (const float*)d_in[7];
  const float* lsd   = (const float*)d_in[8];
  const float* WOw   = (const float*)d_in[9];
  const float* WOb   = (const float*)d_in[10];
  float* out = (float*)d_out;

  float* w = (float*)d_ws;
  auto alloc = [&](size_t n) { float* p = w; w += n; return p; };
  float* q     = alloc((size_t)B * H * L * D);  // becomes scaled q in-place
  float* vg    = alloc((size_t)B * H * L * D);
  float* a2    = alloc((size_t)B * H * L);
  float* kb    = alloc((size_t)H * M * D);      // becomes scaled k_beta in-place
  float* bk2   = alloc((size_t)H * M);
  float* invls = alloc((size_t)H * D);
  float* sfv   = alloc(8);
  float* kkk   = alloc((size_t)H * M * M);
  float* chol  = alloc((size_t)H * M * M);
  float* kqk   = alloc((size_t)B * H * L * M);
  float* v1    = alloc((size_t)B * H * M * L);
  float* rhs2  = alloc((size_t)B * H * M * D);
  float* v2    = alloc((size_t)B * H * M * D);
  float* mean1 = alloc((size_t)B * H * L * D);
  float* meanv = alloc((size_t)B * H * L * D);
  float* sl    = alloc((size_t)H * D * M * M);
  float* G     = alloc((size_t)H * D * M * M);
  float* v3sq  = alloc((size_t)B * H * L * D);
  float* v1sq  = alloc((size_t)B * H * L);
  float* samp  = alloc((size_t)B * S * L * HD);
  float* klp   = alloc(16);

  k_qkv<<<(B * L / 16) * (QKVD / 16) / 4, 128, 0, stream>>>(x, Wqkv, q, vg);
  k_kbeta<<<(M / 16) * (HD / 16) / 4, 128, 0, stream>>>(curk, Wqkv, kb);
  k_expp<<<1, 512, 0, stream>>>(logls, logsf, invls, sfv);
  k_scaleq<<<B * H * L, 64, 0, stream>>>(q, invls, a2);
  k_scalek<<<H * M, 64, 0, stream>>>(kb, invls, bk2);
  k_kkk<<<H * 16 / 4, 128, 0, stream>>>(kb, bk2, sfv, kkk);
  k_chol<<<H, 64, 0, stream>>>(kkk, chol);
  k_kqkb<<<B * H * 128 / 4, 128, 0, stream>>>(q, a2, kb, bk2, sfv, kqk);
  k_mean1<<<B * H * 32 / 4, 128, 0, stream>>>(q, a2, sfv, vg, mean1);
  k_trsv1<<<B * H * 2, 256, 0, stream>>>(chol, kqk, v1);
  k_rhs2<<<B * H * 16 / 4, 128, 0, stream>>>(kqk, vg, rhs2);
  k_trsv2<<<B * H, 64, 0, stream>>>(chol, rhs2, v2);
  k_slocal<<<(H * D * M * M) / 256, 256, 0, stream>>>(lsd, ltri, sl);
  k_G<<<H * D * 16 / 4, 128, 0, stream>>>(sl, G);
  k_v3sq<<<B * H * D * 32 / 4, 128, 0, stream>>>(G, v1, v3sq);
  k_v1sq<<<B * H * 2, 256, 0, stream>>>(v1, v1sq);
  k_mean<<<B * H * 128 / 4, 128, 0, stream>>>(kqk, vind, v1, v2, mean1, meanv);
  k_samples<<<(B * S * L * HD) / 256, 256, 0, stream>>>(meanv, v3sq, v1sq, sfv, eps, samp);
  k_out<<<(B * S * L / 16) * (HD / 16) / 4, 128, 0, stream>>>(samp, WOw, WOb, out);
  k_red_sl<<<1, 256, 0, stream>>>(sl, lsd, klp);
  k_red_vm<<<1, 256, 0, stream>>>(vg, mean1, v2, klp);
  k_term3<<<H, 64, 0, stream>>>(kkk, vind, klp + 8);
  k_fin<<<1, 1, 0, stream>>>(klp, out);
}